// King2019_4148938408292
// MI455X (gfx1250) — compile-verified
//
#include <hip/hip_runtime.h>
#include <math.h>

// ---------------- problem constants ----------------
#define T_LEN   32000
#define NF      31
#define NM      10
#define CHUNK   256      // samples per front-end chunk (16 sub-blocks of 16)
#define NCHUNK  125      // 32000 / 256
#define FS_F    32000.0f

typedef float v2f __attribute__((ext_vector_type(2)));
typedef float v8f __attribute__((ext_vector_type(8)));

#if defined(__has_builtin)
#if __has_builtin(__builtin_amdgcn_wmma_f32_16x16x4_f32)
#define HAVE_WMMA_F32X4 1
#endif
#endif

__device__ __forceinline__ float2 cmul(float2 a, float2 b) {
  return make_float2(__builtin_fmaf(a.x, b.x, -a.y * b.y),
                     __builtin_fmaf(a.x, b.y,  a.y * b.x));
}
__device__ __forceinline__ float2 cadd(float2 a, float2 b) {
  return make_float2(a.x + b.x, a.y + b.y);
}

// =====================================================================
// Kernel 1: gammatone (WMMA block-Toeplitz) + compression + IHC LP + HP
// one wave32 per (b, f) sequence; writes ad[(b*NF+f)*T + t] to workspace
// =====================================================================
__global__ __launch_bounds__(32)
void k_frontend(const float* __restrict__ x, float* __restrict__ ad)
{
  __shared__ float xs[CHUNK];        // chunk staging (x, then HWR(c))
  __shared__ float hre_s[16];        // Re of impulse response h[d]
  __shared__ float G_s[16][8];       // hom-output map: (M^{t+1})_{4,c} re/im
  __shared__ float W_s[16][8];       // state injection map M^{15-k} v re/im
  __shared__ float u_s[16][8];       // per-sub-block injections this chunk

  const int seq  = blockIdx.x;       // b*NF + f
  const int b    = seq / NF;
  const int f    = seq - b * NF;
  const int lane = threadIdx.x;
  const int m    = lane & 15;        // sub-block / matrix row index
  const int hi   = lane >> 4;

  // ---- gammatone channel constants (1-ERB spacing, 80..8000 Hz) ----
  const float e1  = 9.2645f * logf(1.0f + 80.0f   / 228.8455f);
  const float e2  = 9.2645f * logf(1.0f + 8000.0f / 228.8455f);
  const float erb = e1 + (float)f * (e2 - e1) / (float)(NF - 1);
  const float fc  = 228.8455f * (expf(erb / 9.2645f) - 1.0f);
  const float bw  = 1.0186f * (24.7f + fc / 9.265f);
  const float lam = expf(-2.0f * (float)M_PI * bw / FS_F);
  const float th  = 2.0f * (float)M_PI * fc / FS_F;
  const float2 a  = make_float2(lam * cosf(th), lam * sinf(th));
  const float om  = 1.0f - lam;
  const float gain = 2.0f * om * om * om * om;
  float2 a2 = cmul(a, a), a4 = cmul(a2, a2), a8c = cmul(a4, a4);
  const float2 a16 = cmul(a8c, a8c);   // M^16 = a^16 * L^16 (binomial lower-tri)

  // ---- build block-filter tables (cheap, once, lane 0) ----
  if (lane == 0) {
    float2 s[4];
    // impulse response through the 4-pole cascade: h[d] = e4^T M^d v
    s[0] = s[1] = s[2] = s[3] = make_float2(0.f, 0.f);
    for (int d = 0; d < 16; ++d) {
      float2 inp = make_float2(d == 0 ? 1.f : 0.f, 0.f);
      for (int k = 0; k < 4; ++k) { s[k] = cadd(cmul(a, s[k]), inp); inp = s[k]; }
      hre_s[d] = s[3].x;
    }
    // G[t][c] = (M^{t+1})_{4,c}: run zero-input cascade from e_c
    for (int c = 0; c < 4; ++c) {
      s[0] = s[1] = s[2] = s[3] = make_float2(0.f, 0.f);
      s[c] = make_float2(1.f, 0.f);
      for (int t = 0; t < 16; ++t) {
        float2 inp = make_float2(0.f, 0.f);
        for (int k = 0; k < 4; ++k) { s[k] = cadd(cmul(a, s[k]), inp); inp = s[k]; }
        G_s[t][c] = s[3].x; G_s[t][4 + c] = s[3].y;
      }
    }
    // W[k] = M^{15-k} v  (v = ones): iterate backward from v
    s[0] = s[1] = s[2] = s[3] = make_float2(1.f, 0.f);
    for (int k = 15; k >= 0; --k) {
      for (int c = 0; c < 4; ++c) { W_s[k][c] = s[c].x; W_s[k][4 + c] = s[c].y; }
      if (k > 0) {
        float2 inp = make_float2(0.f, 0.f);
        for (int kk = 0; kk < 4; ++kk) { s[kk] = cadd(cmul(a, s[kk]), inp); inp = s[kk]; }
      }
    }
  }
  __syncthreads();

  // ---- A fragments (constant across chunks) ----
  // 16x4 f32 A layout: lanes 0-15 hold M=0..15 K={0,1}; lanes 16-31 K={2,3}
  v2f aY[4], aU[4];
#pragma unroll
  for (int g = 0; g < 4; ++g) {
    const int kk = 4 * g + 2 * hi;
    aY[g].x = (m >= kk    ) ? hre_s[m - kk]     : 0.f;   // Toeplitz Re(h)
    aY[g].y = (m >= kk + 1) ? hre_s[m - kk - 1] : 0.f;
    aU[g].x = (m < 8) ? W_s[kk][m]     : 0.f;            // rows = 8 real comps of u
    aU[g].y = (m < 8) ? W_s[kk + 1][m] : 0.f;
  }

  // ---- stage 2..4 constants ----
  const float tlin = 0.00031622776601683794f;            // 10^((30-100)/20)
  const float Ki = tanf((float)M_PI * 1000.0f / FS_F);
  const float bI = Ki / (1.f + Ki);
  const float mI = -((Ki - 1.f) / (1.f + Ki));           // y = bI(x+xp) + mI*yp
  const float Kh = tanf((float)M_PI * 3.0f / FS_F);
  const float bH = 1.f / (1.f + Kh);
  const float mH = -((Kh - 1.f) / (1.f + Kh));           // y = bH(x-xp) + mH*yp
  const float mI2 = mI * mI, mI4 = mI2 * mI2, mI8 = mI4 * mI4;
  const float mH2 = mH * mH, mH4 = mH2 * mH2, mH8 = mH4 * mH4;

  float2 sst[4];                                         // gammatone state
  sst[0] = sst[1] = sst[2] = sst[3] = make_float2(0.f, 0.f);
  float cx = 0.f, cy = 0.f;                              // IHC carries
  float ex = 0.f, ey = 0.f;                              // HP carries

  const float* xrow  = x  + (size_t)b   * T_LEN;
  float*       adrow = ad + (size_t)seq * T_LEN;

  for (int chn = 0; chn < NCHUNK; ++chn) {
    const int base = chn * CHUNK;
    { // coalesced chunk load -> LDS
      const float4 p0 = *(const float4*)(xrow + base + lane * 8);
      const float4 p1 = *(const float4*)(xrow + base + lane * 8 + 4);
      *(float4*)(xs + lane * 8)     = p0;
      *(float4*)(xs + lane * 8 + 4) = p1;
    }
    __syncthreads();

    // ---- zero-state responses (dY) and state injections (dU) via WMMA ----
    v8f dY = {0.f, 0.f, 0.f, 0.f, 0.f, 0.f, 0.f, 0.f};
    v8f dU = {0.f, 0.f, 0.f, 0.f, 0.f, 0.f, 0.f, 0.f};
#if defined(HAVE_WMMA_F32X4)
#pragma unroll
    for (int g = 0; g < 4; ++g) {
      const int kk = 4 * g + 2 * hi;
      v2f bf; bf.x = xs[16 * m + kk]; bf.y = xs[16 * m + kk + 1]; // B: N=m, rows kk
      dY = __builtin_amdgcn_wmma_f32_16x16x4_f32(false, aY[g], false, bf,
                                                 (short)0, dY, false, false);
      dU = __builtin_amdgcn_wmma_f32_16x16x4_f32(false, aU[g], false, bf,
                                                 (short)0, dU, false, false);
    }
#else
#pragma unroll
    for (int r = 0; r < 8; ++r) {
      const int t = r + 8 * hi;
      float acc = 0.f;
      for (int k = 0; k <= t; ++k) acc = __builtin_fmaf(hre_s[t - k], xs[16 * m + k], acc);
      dY[r] = acc;
      float au = 0.f;
      if (hi == 0) for (int k = 0; k < 16; ++k) au = __builtin_fmaf(W_s[k][r], xs[16 * m + k], au);
      dU[r] = au;
    }
#endif
    if (hi == 0) {                       // publish u_j components
#pragma unroll
      for (int r = 0; r < 8; ++r) u_s[m][r] = dU[r];
    }
    __syncthreads();

    // ---- sub-block state recurrence s_{j+1} = M^16 s_j + u_j (16 steps) ----
    float2 sj[4]  = { sst[0], sst[1], sst[2], sst[3] };
    float2 mys[4] = { sj[0], sj[1], sj[2], sj[3] };
    for (int jj = 0; jj < 16; ++jj) {
      if (jj == m) { mys[0] = sj[0]; mys[1] = sj[1]; mys[2] = sj[2]; mys[3] = sj[3]; }
      const float2 u0 = make_float2(u_s[jj][0], u_s[jj][4]);
      const float2 u1 = make_float2(u_s[jj][1], u_s[jj][5]);
      const float2 u2 = make_float2(u_s[jj][2], u_s[jj][6]);
      const float2 u3 = make_float2(u_s[jj][3], u_s[jj][7]);
      const float2 t1 = sj[0];
      const float2 t2 = make_float2(16.f * sj[0].x + sj[1].x, 16.f * sj[0].y + sj[1].y);
      const float2 t3 = make_float2(136.f * sj[0].x + 16.f * sj[1].x + sj[2].x,
                                    136.f * sj[0].y + 16.f * sj[1].y + sj[2].y);
      const float2 t4 = make_float2(816.f * sj[0].x + 136.f * sj[1].x + 16.f * sj[2].x + sj[3].x,
                                    816.f * sj[0].y + 136.f * sj[1].y + 16.f * sj[2].y + sj[3].y);
      sj[0] = cadd(cmul(a16, t1), u0);
      sj[1] = cadd(cmul(a16, t2), u1);
      sj[2] = cadd(cmul(a16, t3), u2);
      sj[3] = cadd(cmul(a16, t4), u3);
    }
    sst[0] = sj[0]; sst[1] = sj[1]; sst[2] = sj[2]; sst[3] = sj[3];
    __syncthreads();

    // ---- homogeneous term + gain + compression + HWR, to LDS time-major ----
#pragma unroll
    for (int r = 0; r < 8; ++r) {
      const int t = r + 8 * hi;
      float hr = 0.f;
#pragma unroll
      for (int c = 0; c < 4; ++c) {
        hr = __builtin_fmaf( G_s[t][c],     mys[c].x, hr);
        hr = __builtin_fmaf(-G_s[t][4 + c], mys[c].y, hr);
      }
      const float y  = (dY[r] + hr) * gain;
      const float ay = fabsf(y);
      float cval;
      if (ay < tlin) cval = y;
      else cval = copysignf(tlin * __powf(ay * (1.0f / tlin), 0.3f), y);
      xs[16 * m + t] = fmaxf(cval, 0.f);      // HWR for IHC input
    }
    __syncthreads();

    float xv[8];
#pragma unroll
    for (int i = 0; i < 8; ++i) xv[i] = xs[lane * 8 + i];

    // ---- IHC LP: chunked affine scan (constant pole mI) ----
    const float lastx = xv[7];
    float px = __shfl_up(xv[7], 1, 32);
    if (lane == 0) px = cx;
    float w[8];
    { float xp = px, acc = 0.f;
#pragma unroll
      for (int i = 0; i < 8; ++i) {
        acc = __builtin_fmaf(mI, acc, bI * (xv[i] + xp));
        w[i] = acc; xp = xv[i];
      } }
    float P = mI8, Q = w[7];
#pragma unroll
    for (int d2 = 1; d2 < 32; d2 <<= 1) {
      const float Pu = __shfl_up(P, d2, 32);
      const float Qu = __shfl_up(Q, d2, 32);
      if (lane >= d2) { Q = __builtin_fmaf(P, Qu, Q); P = P * Pu; }
    }
    const float yend = __builtin_fmaf(P, cy, Q);
    float yin = __shfl_up(yend, 1, 32);
    if (lane == 0) yin = cy;
    float evv[8];
    { float mp = mI;
#pragma unroll
      for (int i = 0; i < 8; ++i) { evv[i] = __builtin_fmaf(mp, yin, w[i]); mp *= mI; } }
    cx = __shfl(lastx, 31, 32);
    cy = __shfl(yend, 31, 32);

    // ---- HP: same scan structure ----
    const float laste = evv[7];
    float pe = __shfl_up(evv[7], 1, 32);
    if (lane == 0) pe = ex;
    float wh[8];
    { float ep = pe, acc = 0.f;
#pragma unroll
      for (int i = 0; i < 8; ++i) {
        acc = __builtin_fmaf(mH, acc, bH * (evv[i] - ep));
        wh[i] = acc; ep = evv[i];
      } }
    float Ph = mH8, Qh = wh[7];
#pragma unroll
    for (int d2 = 1; d2 < 32; d2 <<= 1) {
      const float Pu = __shfl_up(Ph, d2, 32);
      const float Qu = __shfl_up(Qh, d2, 32);
      if (lane >= d2) { Qh = __builtin_fmaf(Ph, Qu, Qh); Ph = Ph * Pu; }
    }
    const float hend = __builtin_fmaf(Ph, ey, Qh);
    float hin = __shfl_up(hend, 1, 32);
    if (lane == 0) hin = ey;
    float adv[8];
    { float mp = mH;
#pragma unroll
      for (int i = 0; i < 8; ++i) { adv[i] = __builtin_fmaf(mp, hin, wh[i]); mp *= mH; } }
    ex = __shfl(laste, 31, 32);
    ey = __shfl(hend, 31, 32);

    *(float4*)(adrow + base + lane * 8)     = make_float4(adv[0], adv[1], adv[2], adv[3]);
    *(float4*)(adrow + base + lane * 8 + 4) = make_float4(adv[4], adv[5], adv[6], adv[7]);
    __syncthreads();
  }
}

// ---------------- modulation filterbank coefficients ----------------
__device__ __forceinline__ void mod_coefs(int mi, float* b0, float* b2,
                                          float* a1c, float* a2c)
{
  const float lg2 = 0.301029995663981f, lg150 = 2.176091259055681f;
  const float mfc = __powf(10.f, lg2 + (float)mi * (lg150 - lg2) / 9.f);
  const float w0 = 2.f * (float)M_PI * mfc / FS_F;
  const float alpha = sinf(w0) * 0.5f;        // Q = 1
  const float a0 = 1.f + alpha;
  *b0  =  alpha / a0;
  *b2  = -alpha / a0;
  *a1c = -2.f * cosf(w0) / a0;
  *a2c = (1.f - alpha) / a0;
}

// =====================================================================
// Kernel 2: per-(b,f,m) biquad chunk-boundary states via 2x2-matrix scan
// one wave32 per sequence; writes z at every chunk start to zs
// =====================================================================
__global__ __launch_bounds__(32)
void k_modstate(const float* __restrict__ ad, float* __restrict__ zs)
{
  const int seq  = blockIdx.x;          // (b*NF+f)*NM + m
  const int mi   = seq % NM;
  const int bf   = seq / NM;
  const int lane = threadIdx.x;
  float b0, b2, a1c, a2c; mod_coefs(mi, &b0, &b2, &a1c, &a2c);
  // z' = A z + wv x :  A = [[-a1,1],[-a2,0]], wv = (-a1*b0, b2 - a2*b0)
  const float A00 = -a1c, A01 = 1.f, A10 = -a2c;
  const float w1 = -a1c * b0, w2v = b2 - a2c * b0;
  float4 Ap = make_float4(A00, A01, A10, 0.f);   // A^8 by squaring
#pragma unroll
  for (int i = 0; i < 3; ++i) {
    const float4 Bp = Ap;
    Ap = make_float4(Bp.x * Bp.x + Bp.y * Bp.z, Bp.x * Bp.y + Bp.y * Bp.w,
                     Bp.z * Bp.x + Bp.w * Bp.z, Bp.z * Bp.y + Bp.w * Bp.w);
  }
  float z0x = 0.f, z0y = 0.f;
  const float* xrow = ad + (size_t)bf * T_LEN;
  for (int chn = 0; chn < NCHUNK; ++chn) {
    if (lane == 0) {
      zs[((size_t)seq * NCHUNK + chn) * 2]     = z0x;
      zs[((size_t)seq * NCHUNK + chn) * 2 + 1] = z0y;
    }
    const float4 p0 = *(const float4*)(xrow + chn * CHUNK + lane * 8);
    const float4 p1 = *(const float4*)(xrow + chn * CHUNK + lane * 8 + 4);
    const float xv[8] = {p0.x, p0.y, p0.z, p0.w, p1.x, p1.y, p1.z, p1.w};
    float dx = 0.f, dy = 0.f;                   // lane-local zero-state
#pragma unroll
    for (int i = 0; i < 8; ++i) {
      const float nx = __builtin_fmaf(A00, dx, __builtin_fmaf(A01, dy, w1 * xv[i]));
      const float ny = __builtin_fmaf(A10, dx, w2v * xv[i]);
      dx = nx; dy = ny;
    }
    float4 P = Ap; float qx = dx, qy = dy;      // inclusive scan of (A^8, d)
#pragma unroll
    for (int d2 = 1; d2 < 32; d2 <<= 1) {
      const float Px = __shfl_up(P.x, d2, 32), Py = __shfl_up(P.y, d2, 32);
      const float Pz = __shfl_up(P.z, d2, 32), Pw = __shfl_up(P.w, d2, 32);
      const float qxu = __shfl_up(qx, d2, 32), qyu = __shfl_up(qy, d2, 32);
      if (lane >= d2) {
        const float nqx = __builtin_fmaf(P.x, qxu, __builtin_fmaf(P.y, qyu, qx));
        const float nqy = __builtin_fmaf(P.z, qxu, __builtin_fmaf(P.w, qyu, qy));
        P = make_float4(P.x * Px + P.y * Pz, P.x * Py + P.y * Pw,
                        P.z * Px + P.w * Pz, P.z * Py + P.w * Pw);
        qx = nqx; qy = nqy;
      }
    }
    const float zex = __builtin_fmaf(P.x, z0x, __builtin_fmaf(P.y, z0y, qx));
    const float zey = __builtin_fmaf(P.z, z0x, __builtin_fmaf(P.w, z0y, qy));
    z0x = __shfl(zex, 31, 32);
    z0y = __shfl(zey, 31, 32);
  }
}

// =====================================================================
// Kernel 3: parallel-in-time writeout. block = (b, chunk); thread = (f,m).
// Stages y in LDS tiles and writes fully coalesced [t x 310] slabs.
// =====================================================================
#define SUBT 32
__global__ __launch_bounds__(320)
void k_modwrite(const float* __restrict__ ad, const float* __restrict__ zs,
                float* __restrict__ out)
{
  __shared__ float tile[SUBT * (NF * NM)];      // 32*310*4B = 39.7 KB
  const int bi  = blockIdx.x / NCHUNK;
  const int chn = blockIdx.x % NCHUNK;
  const int tid = threadIdx.x;
  const int fi = tid / NM, mi = tid % NM;
  const bool act = (tid < NF * NM);
  float b0, b2, a1c, a2c; mod_coefs(mi, &b0, &b2, &a1c, &a2c);
  float z1 = 0.f, z2 = 0.f;
  const float* xrow = ad;
  if (act) {
    const int seq = (bi * NF + fi) * NM + mi;
    z1 = zs[((size_t)seq * NCHUNK + chn) * 2];
    z2 = zs[((size_t)seq * NCHUNK + chn) * 2 + 1];
    xrow = ad + (size_t)(bi * NF + fi) * T_LEN + chn * CHUNK;
  }
  for (int sub = 0; sub < CHUNK / SUBT; ++sub) {
    if (act) {
#pragma unroll 4
      for (int tt = 0; tt < SUBT; ++tt) {
        const float xvv = xrow[sub * SUBT + tt];
        const float y   = __builtin_fmaf(b0, xvv, z1);
        const float z1n = __builtin_fmaf(-a1c, y, z2);
        const float z2n = __builtin_fmaf(b2, xvv, -a2c * y);
        z1 = z1n; z2 = z2n;
        tile[tt * (NF * NM) + tid] = y;
      }
    }
    __syncthreads();
    const size_t obase =
        ((size_t)bi * T_LEN + (size_t)chn * CHUNK + (size_t)sub * SUBT) * (size_t)(NF * NM);
    for (int idx = tid; idx < SUBT * NF * NM; idx += 320) out[obase + idx] = tile[idx];
    __syncthreads();
  }
}

// =====================================================================
extern "C" void kernel_launch(void* const* d_in, const int* in_sizes, int n_in,
                              void* d_out, int out_size, void* d_ws, size_t ws_size,
                              hipStream_t stream)
{
  (void)n_in; (void)out_size; (void)ws_size;
  const float* x = (const float*)d_in[0];
  int B = in_sizes[0] / T_LEN;
  if (B <= 0) B = 1;

  float* ws  = (float*)d_ws;
  float* adw = ws;                                   // [B*NF, T] f32
  const size_t adN = (size_t)B * NF * T_LEN;
  float* zsw = ws + adN;                             // [B*NF*NM, NCHUNK, 2] f32

  k_frontend<<<dim3(B * NF),      dim3(32),  0, stream>>>(x, adw);
  k_modstate<<<dim3(B * NF * NM), dim3(32),  0, stream>>>(adw, zsw);
  k_modwrite<<<dim3(B * NCHUNK),  dim3(320), 0, stream>>>(adw, zsw, (float*)d_out);
}